// FuzzyAttentionLayer_90941637525592
// MI455X (gfx1250) — compile-verified
//
#include <hip/hip_runtime.h>

typedef __attribute__((ext_vector_type(2))) float v2f;
typedef __attribute__((ext_vector_type(8))) float v8f;

#define IN_CH 256
#define HEADS 4
#define CPH   64
#define RULES 8
#define NEG_SLOPE 0.2f

// ---------------------------------------------------------------------------
// 0) zero scratch (mkey / denom / agg are laid out contiguously)
// ---------------------------------------------------------------------------
__global__ void init_zero(unsigned* __restrict__ p, size_t n) {
    size_t i = (size_t)blockIdx.x * blockDim.x + threadIdx.x;
    size_t stride = (size_t)gridDim.x * blockDim.x;
    for (; i < n; i += stride) p[i] = 0u;
}

// ---------------------------------------------------------------------------
// 1) xp = x @ W   (f32 GEMM via V_WMMA_F32_16X16X4_F32)
//    Block b owns a 16-row A panel (rows (b>>1)*16 ..+16); its 8 waves cover
//    8 consecutive 16-col N-tiles ((b&1)*8 + wave).  The shared A panel
//    (16x256 f32 = 16KB) is staged ONCE per block into LDS with
//    GLOBAL_LOAD_ASYNC_TO_LDS_B128 (ASYNCcnt path), then each wave reads its
//    A fragments from LDS (ds_load_b64) inside the K loop.  B stays in
//    global (L2-resident, 256KB total).
// ---------------------------------------------------------------------------
__global__ __launch_bounds__(256) void proj_wmma(const float* __restrict__ x,
                                                 const float* __restrict__ W,
                                                 float* __restrict__ xp) {
    __shared__ __align__(16) float As[16 * IN_CH];   // 16KB A panel

    const int tid  = threadIdx.x;
    const int wave = tid >> 5;
    const int lane = tid & 31;
    const int half = lane >> 4;     // lane group: 0 -> K=(k0,k0+1), 1 -> (k0+2,k0+3)
    const int l16  = lane & 15;

    const int mBase = (blockIdx.x >> 1) * 16;
    const int nBase = (((blockIdx.x & 1) << 3) + wave) * 16;

    // ---- stage A panel: 1024 x 16B chunks, 256 threads x 4 iterations ----
    {
        const float*   gA     = x + (size_t)mBase * IN_CH;   // 16 rows x 256 cols
        const unsigned ldsTop = (unsigned)(uintptr_t)(&As[0]);
#pragma unroll
        for (int it = 0; it < 4; ++it) {
            const int      chunk  = it * 256 + tid;          // 16-byte chunk id
            const unsigned ldsOff = ldsTop + chunk * 16;
            const float*   gPtr   = gA + chunk * 4;
            asm volatile("global_load_async_to_lds_b128 %0, %1, off"
                         :: "v"(ldsOff), "v"(gPtr)
                         : "memory");
        }
        asm volatile("s_wait_asynccnt 0x0" ::: "memory");
    }
    __syncthreads();

    const float* Bcol = W + nBase + l16;                    // B: col = l16

    v8f acc = {};
#pragma unroll 4
    for (int k0 = 0; k0 < IN_CH; k0 += 4) {
        const int k = k0 + half * 2;
        v2f a = *(const v2f*)(&As[l16 * IN_CH + k]);        // ds_load_b64
        v2f b;
        b.x = Bcol[(size_t)k * IN_CH];                      // B[k,   col]
        b.y = Bcol[(size_t)(k + 1) * IN_CH];                // B[k+1, col]
        acc = __builtin_amdgcn_wmma_f32_16x16x4_f32(
            /*neg_a=*/false, a, /*neg_b=*/false, b,
            /*c_mod=*/(short)0, acc, /*reuse_a=*/false, /*reuse_b=*/false);
    }

    // C/D layout: VGPR v, lane -> (M = v + 8*half, N = l16)
    float* Crow = xp + (size_t)(mBase + half * 8) * IN_CH + nBase + l16;
#pragma unroll
    for (int v = 0; v < 8; ++v) Crow[(size_t)v * IN_CH] = acc[v];
}

// ---------------------------------------------------------------------------
// 2) per-(node,head) attention scalars: a_src/a_dst = <xp[n,h,:], att_*[h,:]>
// ---------------------------------------------------------------------------
__global__ void att_scalars(const float* __restrict__ xp,
                            const float* __restrict__ att_src,
                            const float* __restrict__ att_dst,
                            float* __restrict__ a_s, float* __restrict__ a_d,
                            int nNodes) {
    int i = blockIdx.x * blockDim.x + threadIdx.x;   // i = n*HEADS + h
    if (i >= nNodes * HEADS) return;
    int h = i & (HEADS - 1);
    const float* row = xp + (size_t)(i >> 2) * IN_CH + h * CPH;
    const float* as  = att_src + h * CPH;
    const float* ad  = att_dst + h * CPH;
    float s = 0.f, d = 0.f;
#pragma unroll 4
    for (int c = 0; c < CPH; ++c) { float v = row[c]; s += v * as[c]; d += v * ad[c]; }
    a_s[i] = s; a_d[i] = d;
}

__device__ __forceinline__ float leaky(float l) {
    return l > 0.f ? l : NEG_SLOPE * l;
}
// order-preserving float -> uint mapping for atomicMax-based segment max
__device__ __forceinline__ unsigned fkey(float f) {
    unsigned b = __float_as_uint(f);
    return (b & 0x80000000u) ? ~b : (b | 0x80000000u);
}
__device__ __forceinline__ float funkey(unsigned k) {
    return (k & 0x80000000u) ? __uint_as_float(k & 0x7FFFFFFFu)
                             : __uint_as_float(~k);
}

// ---------------------------------------------------------------------------
// 3) edge logits + segment max into mkey[dst,h] (self loops appended)
// ---------------------------------------------------------------------------
__global__ void edge_max(const int* __restrict__ src, const int* __restrict__ dst,
                         const float* __restrict__ a_s, const float* __restrict__ a_d,
                         unsigned* __restrict__ mkey, int nE, int nN) {
    int e = blockIdx.x * blockDim.x + threadIdx.x;
    if (e >= nE + nN) return;
    int s, d;
    if (e < nE) { s = src[e]; d = dst[e]; } else { s = d = e - nE; }
#pragma unroll
    for (int h = 0; h < HEADS; ++h) {
        float l = leaky(a_s[s * HEADS + h] + a_d[d * HEADS + h]);
        atomicMax(&mkey[d * HEADS + h], fkey(l));
    }
}

// ---------------------------------------------------------------------------
// 4) scatter: p = exp(l - m[dst]); denom[dst,h] += p; agg[dst,:] += p*xp[src,:]
//    one 256-thread block per edge; thread t owns channel t (head t/64).
//    Normalization (÷denom) deferred to the epilogue.
// ---------------------------------------------------------------------------
__global__ __launch_bounds__(256) void edge_scatter(
        const int* __restrict__ src, const int* __restrict__ dst,
        const float* __restrict__ a_s, const float* __restrict__ a_d,
        const unsigned* __restrict__ mkey, const float* __restrict__ xp,
        float* __restrict__ denom, float* __restrict__ agg, int nE) {
    int e = blockIdx.x;
    int t = threadIdx.x;            // channel 0..255, head = t>>6
    int s, d;
    if (e < nE) { s = src[e]; d = dst[e]; } else { s = d = e - nE; }
    int h = t >> 6;
    float l = leaky(a_s[s * HEADS + h] + a_d[d * HEADS + h]);
    float m = funkey(mkey[d * HEADS + h]);
    float p = __expf(l - m);
    if ((t & 63) == 0) atomicAdd(&denom[d * HEADS + h], p);
    atomicAdd(&agg[(size_t)d * IN_CH + t], p * xp[(size_t)s * IN_CH + t]);
}

// ---------------------------------------------------------------------------
// 5) fuzzy-gate hidden layer: hidden[n,r] = sigmoid(x[n,:] @ gw1[:,r] + gb1[r])
// ---------------------------------------------------------------------------
__global__ void gate_hidden(const float* __restrict__ x,
                            const float* __restrict__ gw1,
                            const float* __restrict__ gb1,
                            float* __restrict__ hidden, int nN) {
    int i = blockIdx.x * blockDim.x + threadIdx.x;   // n*RULES + r
    if (i >= nN * RULES) return;
    int r = i & (RULES - 1);
    const float* row = x + (size_t)(i >> 3) * IN_CH;
    float acc = gb1[r];
#pragma unroll 4
    for (int k = 0; k < IN_CH; ++k) acc += row[k] * gw1[k * RULES + r];
    hidden[i] = 1.f / (1.f + __expf(-acc));
}

// ---------------------------------------------------------------------------
// 6) epilogue: head-mean of normalized agg + bias, times sigmoid gate
// ---------------------------------------------------------------------------
__global__ void finalize(const float* __restrict__ agg, const float* __restrict__ denom,
                         const float* __restrict__ bias, const float* __restrict__ hidden,
                         const float* __restrict__ gw2, const float* __restrict__ gb2,
                         float* __restrict__ out, int nN) {
    int i = blockIdx.x * blockDim.x + threadIdx.x;   // n*CPH + c
    if (i >= nN * CPH) return;
    int n = i >> 6, c = i & (CPH - 1);
    float sum = 0.f;
#pragma unroll
    for (int h = 0; h < HEADS; ++h)
        sum += agg[(size_t)n * IN_CH + h * CPH + c] / denom[n * HEADS + h];
    float attn = sum * (1.f / HEADS) + bias[c];
    float z = gb2[c];
#pragma unroll
    for (int r = 0; r < RULES; ++r) z += hidden[n * RULES + r] * gw2[r * CPH + c];
    float gate = 1.f / (1.f + __expf(-z));
    out[i] = attn * gate;
}

// ---------------------------------------------------------------------------
extern "C" void kernel_launch(void* const* d_in, const int* in_sizes, int n_in,
                              void* d_out, int out_size, void* d_ws, size_t ws_size,
                              hipStream_t stream) {
    (void)n_in; (void)out_size; (void)ws_size;
    const float* x    = (const float*)d_in[0];
    const int*   ei   = (const int*)d_in[1];
    const float* W    = (const float*)d_in[2];
    const float* asrc = (const float*)d_in[3];
    const float* adst = (const float*)d_in[4];
    const float* bias = (const float*)d_in[5];
    const float* gw1  = (const float*)d_in[6];
    const float* gb1  = (const float*)d_in[7];
    const float* gw2  = (const float*)d_in[8];
    const float* gb2  = (const float*)d_in[9];
    float* out = (float*)d_out;

    const int nN = in_sizes[0] / IN_CH;   // 50000
    const int nE = in_sizes[1] / 2;       // 800000
    const int* src = ei;
    const int* dst = ei + nE;

    // workspace layout (floats)
    float*    xp     = (float*)d_ws;                         // nN*256
    float*    a_s    = xp  + (size_t)nN * IN_CH;             // nN*4
    float*    a_d    = a_s + (size_t)nN * HEADS;             // nN*4
    unsigned* mkey   = (unsigned*)(a_d + (size_t)nN * HEADS);// nN*4
    float*    denom  = (float*)(mkey + (size_t)nN * HEADS);  // nN*4
    float*    agg    = denom + (size_t)nN * HEADS;           // nN*256
    float*    hidden = agg + (size_t)nN * IN_CH;             // nN*8

    // mkey, denom, agg are contiguous: zero them in one pass
    size_t zwords = (size_t)nN * (HEADS + HEADS + IN_CH);
    hipLaunchKernelGGL(init_zero, dim3(2048), dim3(256), 0, stream, mkey, zwords);

    // 1) projection GEMM (WMMA f32 + async-to-LDS A panel):
    //    (nN/16) row panels x 2 blocks (8 N-tiles each)
    hipLaunchKernelGGL(proj_wmma, dim3((nN / 16) * 2), dim3(256), 0, stream, x, W, xp);

    // 2) attention scalars
    int t2 = nN * HEADS;
    hipLaunchKernelGGL(att_scalars, dim3((t2 + 255) / 256), dim3(256), 0, stream,
                       xp, asrc, adst, a_s, a_d, nN);

    // 3) segment max over incoming edges
    int t3 = nE + nN;
    hipLaunchKernelGGL(edge_max, dim3((t3 + 255) / 256), dim3(256), 0, stream,
                       src, dst, a_s, a_d, mkey, nE, nN);

    // 4) exp + unnormalized scatter (one block per edge incl. self loops)
    hipLaunchKernelGGL(edge_scatter, dim3(nE + nN), dim3(256), 0, stream,
                       src, dst, a_s, a_d, mkey, xp, denom, agg, nE);

    // 5) gate hidden layer
    int t5 = nN * RULES;
    hipLaunchKernelGGL(gate_hidden, dim3((t5 + 255) / 256), dim3(256), 0, stream,
                       x, gw1, gb1, hidden, nN);

    // 6) normalize, head-mean, bias, gate
    int t6 = nN * CPH;
    hipLaunchKernelGGL(finalize, dim3((t6 + 255) / 256), dim3(256), 0, stream,
                       agg, denom, bias, hidden, gw2, gb2, out, nN);
}